// Downsampling_53584011985401
// MI455X (gfx1250) — compile-verified
//
#include <hip/hip_runtime.h>

// Problem constants (reference: x = (8, 8192, 512) f32, SAMPLING_FACTOR = 4)
#define B_   8
#define L_   8192
#define D_   512
#define SF   4
#define OUTL (L_ / SF)      // 2048 output rows
#define CH   16             // output rows per tile/chunk
#define CHL  (CH * SF)      // 64 input L-rows per chunk
#define NC   (OUTL / CH)    // 128 chunks along L
#define NDT  (D_ / 16)      // 32 d-tiles of width 16

typedef __attribute__((ext_vector_type(2))) float v2f;
typedef __attribute__((ext_vector_type(8))) float v8f;

// ---------------------------------------------------------------------------
// Phase 1: chunk sums  S[b, c, d] = sum_{i<64} x[b, 64c+i, d]
// One thread per (b, c, d); d fastest -> fully coalesced row reads.
// ---------------------------------------------------------------------------
__global__ void __launch_bounds__(256)
ds_phase1_chunk_sums(const float* __restrict__ x, float* __restrict__ S) {
    int idx = blockIdx.x * 256 + threadIdx.x;       // 8*128*512 = 524288 threads
    int d = idx & (D_ - 1);
    int c = (idx >> 9) & (NC - 1);
    int b = idx >> 16;                              // 9 + 7 bits
    const float* p = x + ((size_t)b * L_ + (size_t)c * CHL) * D_ + d;
    float acc = 0.0f;
#pragma unroll 8
    for (int i = 0; i < CHL; ++i)
        acc += p[(size_t)i * D_];
    S[idx] = acc;
}

// ---------------------------------------------------------------------------
// Phase 2: in-place inclusive scan of S over c, per (b, d) column.
// 4096 independent columns, one thread each (tiny kernel, L2-resident data).
// ---------------------------------------------------------------------------
__global__ void __launch_bounds__(256)
ds_phase2_scan(float* __restrict__ S) {
    int idx = blockIdx.x * 256 + threadIdx.x;       // 8*512 = 4096 threads
    int d = idx & (D_ - 1);
    int b = idx >> 9;
    float* p = S + ((size_t)b * NC) * D_ + d;
    float acc = 0.0f;
    for (int c = 0; c < NC; ++c) {
        acc += p[(size_t)c * D_];
        p[(size_t)c * D_] = acc;                    // inclusive scan (same thread RMW)
    }
}

// ---------------------------------------------------------------------------
// Phase 3: per-tile sampled cumsum via V_WMMA_F32_16X16X4_F32.
//   Tile = 16 output rows (64 L-rows) x 16 d columns, one wave per tile.
//   A (16x4 f32): M = d (one per lane, lanes 0-15 / 16-31 hold K pairs 0,1 / 2,3)
//   B (4x16 f32): constant lower-triangular blocks, synthesized per lane
//   C/D (16x16 f32): lane = output row j, VGPR r = d; seeded with chunk base.
// ---------------------------------------------------------------------------
__global__ void __launch_bounds__(256)
ds_phase3_wmma_scan(const float* __restrict__ x, const float* __restrict__ S,
                    float* __restrict__ out) {
    const int lane = threadIdx.x & 31;
    const int wave = threadIdx.x >> 5;
    const int tile = blockIdx.x * 8 + wave;         // 32768 tiles total

    const int dt   = tile & (NDT - 1);
    const int c    = (tile >> 5) & (NC - 1);
    const int b    = tile >> 12;                    // 5 + 7 bits
    const int db   = dt * 16;
    const int n    = lane & 15;                     // column within tile
    const int half = lane >> 4;                     // 0: lanes 0-15, 1: lanes 16-31

    // Seed accumulator with exclusive chunk base (inclusive scan of previous chunk).
    // VGPR r of C/D holds d = db + r + 8*half for this lane. Two b128 loads.
    v8f acc = {};
    if (c != 0) {
        const float4* bp = (const float4*)
            (S + ((size_t)b * NC + (c - 1)) * D_ + db + 8 * half);
        float4 b0 = bp[0];
        float4 b1 = bp[1];
        acc[0] = b0.x; acc[1] = b0.y; acc[2] = b0.z; acc[3] = b0.w;
        acc[4] = b1.x; acc[5] = b1.y; acc[6] = b1.z; acc[7] = b1.w;
    }

    // Four K=4 WMMAs: A_m[d,k] = g[16c + 4m + k][d],  B_m[k,j] = (4m+k <= j)
    const int k0 = 2 * half;                        // K rows owned by this lane half
#pragma unroll
    for (int m = 0; m < 4; ++m) {
        const int lbase = c * CHL + 16 * m + 8 * half;   // 8 consecutive L rows
        const float* xp = x + ((size_t)b * L_ + lbase) * D_ + db + n;
        float s0 = xp[0 * D_] + xp[1 * D_] + xp[2 * D_] + xp[3 * D_];
        float s1 = xp[4 * D_] + xp[5 * D_] + xp[6 * D_] + xp[7 * D_];
        v2f a;  a.x = s0;  a.y = s1;

        v2f bm;
        bm.x = (4 * m + k0     <= n) ? 1.0f : 0.0f;
        bm.y = (4 * m + k0 + 1 <= n) ? 1.0f : 0.0f;

        acc = __builtin_amdgcn_wmma_f32_16x16x4_f32(
            /*neg_a=*/false, a, /*neg_b=*/false, bm,
            /*c_mod=*/(short)0, acc, /*reuse_a=*/false, /*reuse_b=*/false);
    }

    // Scale by 1/(4*(J+1)) via one v_rcp_f32 (exact-range integers, ~1 ulp),
    // avoiding the full IEEE v_div_scale/v_div_fmas/v_div_fixup sequence.
    const int J = c * CH + n;
    const float scale = 0.25f * __builtin_amdgcn_rcpf((float)(J + 1));

    float* op = out + ((size_t)b * OUTL + J) * D_ + db + 8 * half;
    float4 lo, hi;
    lo.x = acc[0] * scale; lo.y = acc[1] * scale;
    lo.z = acc[2] * scale; lo.w = acc[3] * scale;
    hi.x = acc[4] * scale; hi.y = acc[5] * scale;
    hi.z = acc[6] * scale; hi.w = acc[7] * scale;
    *(float4*)(op)     = lo;
    *(float4*)(op + 4) = hi;
}

// ---------------------------------------------------------------------------
extern "C" void kernel_launch(void* const* d_in, const int* in_sizes, int n_in,
                              void* d_out, int out_size, void* d_ws, size_t ws_size,
                              hipStream_t stream) {
    const float* x = (const float*)d_in[0];
    float* out = (float*)d_out;
    float* S   = (float*)d_ws;   // 8*128*512 f32 = 2 MiB chunk-sum workspace

    // Phase 1: 524288 threads -> 2048 blocks
    hipLaunchKernelGGL(ds_phase1_chunk_sums, dim3((B_ * NC * D_) / 256), dim3(256),
                       0, stream, x, S);
    // Phase 2: 4096 threads -> 16 blocks
    hipLaunchKernelGGL(ds_phase2_scan, dim3((B_ * D_) / 256), dim3(256),
                       0, stream, S);
    // Phase 3: 32768 tiles, 8 waves (tiles) per 256-thread block -> 4096 blocks
    hipLaunchKernelGGL(ds_phase3_wmma_scan, dim3((B_ * NC * NDT) / 8), dim3(256),
                       0, stream, x, S, out);
}